// RGCNEncoder_22582938042963
// MI455X (gfx1250) — compile-verified
//
#include <hip/hip_runtime.h>
#include <hip/hip_bf16.h>

// ---------------------------------------------------------------------------
// RGCN encoder for MI455X (gfx1250, wave32).
//   D = 16, R = 40.  Dense self-loop GEMM uses V_WMMA_F32_16X16X4_F32.
//   Edge message path: 16 lanes per edge, W in LDS, h[src] gathered from L2,
//   per-(dst,rel) mean via a counts histogram, scatter with global f32 atomics
//   (all destination data is L2-resident: 12.8 MB << 192 MB L2).
// ---------------------------------------------------------------------------

#define EMB_DIM 16
#define NUM_RELS 40

typedef __attribute__((ext_vector_type(2))) float v2f;
typedef __attribute__((ext_vector_type(8))) float v8f;

// h0[n][d] = emb[x[n]][d]
__global__ void gather_rows_kernel(const float* __restrict__ emb,
                                   const int* __restrict__ x,
                                   float* __restrict__ h0, int N) {
  int t = blockIdx.x * blockDim.x + threadIdx.x;
  if (t < N * EMB_DIM) {
    int n = t >> 4, d = t & 15;
    h0[t] = emb[x[n] * EMB_DIM + d];
  }
}

__global__ void zero_u32_kernel(unsigned* __restrict__ p, int n) {
  int t = blockIdx.x * blockDim.x + threadIdx.x;
  if (t < n) p[t] = 0u;
}

// counts[dst*R + rel] += 1  (identical for both layers -> computed once)
__global__ void count_edges_kernel(const int* __restrict__ edst,
                                   const int* __restrict__ etype,
                                   unsigned* __restrict__ counts, int E) {
  int e = blockIdx.x * blockDim.x + threadIdx.x;
  if (e < E) atomicAdd(&counts[edst[e] * NUM_RELS + etype[e]], 1u);
}

// out[n][o] = sum_d H[n][d]*root[d][o] + bias[o]
// One wave = one 16-row tile; K=16 done as 4x V_WMMA_F32_16X16X4_F32.
// A layout (16x4 f32): lanes 0-15 hold row M=lane, K={k0,k0+1};
//                      lanes 16-31 hold row M=lane-16, K={k0+2,k0+3}.
// B layout (4x16 f32): lanes 0-15 hold cols N=lane of rows K={k0,k0+1};
//                      lanes 16-31 cols N=lane-16 of rows K={k0+2,k0+3}.
// C/D layout (16x16 f32): VGPR v, lanes 0-15 -> (M=v, N=lane);
//                          lanes 16-31 -> (M=8+v, N=lane-16).
__global__ void root_gemm_wmma_kernel(const float* __restrict__ H,
                                      const float* __restrict__ root,
                                      const float* __restrict__ bias,
                                      float* __restrict__ out, int N) {
  const int gwave = (blockIdx.x * blockDim.x + threadIdx.x) >> 5;
  const int lane  = threadIdx.x & 31;
  const int row0  = gwave << 4;
  if (row0 >= N) return;          // wave-uniform: EXEC all-ones inside
  const int half = lane >> 4;     // 0 or 1
  const int lm   = lane & 15;

  v8f c = {};
  const float* hrow = H + (size_t)(row0 + lm) * EMB_DIM;
#pragma unroll
  for (int k0 = 0; k0 < EMB_DIM; k0 += 4) {
    const int ka = k0 + half * 2;
    v2f a, b;
    a.x = hrow[ka + 0];
    a.y = hrow[ka + 1];
    b.x = root[(ka + 0) * EMB_DIM + lm];
    b.y = root[(ka + 1) * EMB_DIM + lm];
    // 8 args: (neg_a, A, neg_b, B, c_mod, C, reuse_a, reuse_b)
    c = __builtin_amdgcn_wmma_f32_16x16x4_f32(false, a, false, b,
                                              (short)0, c, false, false);
  }
  const float bo = bias[lm];
  const int rbase = row0 + half * 8;
#pragma unroll
  for (int m = 0; m < 8; ++m)
    out[(size_t)(rbase + m) * EMB_DIM + lm] = c[m] + bo;
}

// Edge messages: 16 lanes own one edge; lane o computes output element o.
//   acc_o = sum_d W[r][d][o] * h[src][d]   (h[src] broadcast via __shfl/16)
//   out[dst][o] += acc_o / max(counts[dst*R+r], 1)
// Persistent blocks (grid-stride) so the 40 KB LDS copy of W amortizes.
__global__ void edge_msg_kernel(const float* __restrict__ H,
                                const float* __restrict__ Wg,
                                const unsigned* __restrict__ counts,
                                const int* __restrict__ esrc,
                                const int* __restrict__ edst,
                                const int* __restrict__ etype,
                                float* __restrict__ out, int E) {
  __shared__ float Ws[NUM_RELS * EMB_DIM * EMB_DIM];  // 40 KB
  for (int i = threadIdx.x; i < NUM_RELS * EMB_DIM * EMB_DIM; i += blockDim.x)
    Ws[i] = Wg[i];
  __syncthreads();

  const int o = threadIdx.x & 15;
  const long long total  = (long long)E << 4;
  const long long stride = (long long)gridDim.x * blockDim.x;
  for (long long t = (long long)blockIdx.x * blockDim.x + threadIdx.x;
       t < total; t += stride) {
    const int e = (int)(t >> 4);
    const int src = esrc[e];
    const int dst = edst[e];
    const int r   = etype[e];
    // coalesced 64B gather of one h row across the 16-lane group (L2 hit)
    const float v = H[src * EMB_DIM + o];
    const float* Wr = &Ws[r << 8];
    float acc = 0.f;
#pragma unroll
    for (int d = 0; d < EMB_DIM; ++d) {
      const float hd = __shfl(v, d, 16);   // broadcast within the 16-lane half
      acc = __builtin_fmaf(Wr[(d << 4) + o], hd, acc);
    }
    const float cnt  = (float)counts[dst * NUM_RELS + r]; // uniform in group
    const float norm = 1.0f / fmaxf(cnt, 1.0f);
    atomicAdd(&out[(dst << 4) + o], acc * norm);          // L2 atomic
  }
}

__global__ void relu_kernel(float* __restrict__ p, int n) {
  int t = blockIdx.x * blockDim.x + threadIdx.x;
  if (t < n) p[t] = fmaxf(p[t], 0.0f);
}

extern "C" void kernel_launch(void* const* d_in, const int* in_sizes, int n_in,
                              void* d_out, int out_size, void* d_ws, size_t ws_size,
                              hipStream_t stream) {
  const int*   x     = (const int*)d_in[0];
  const int*   eidx  = (const int*)d_in[1];
  const int*   etype = (const int*)d_in[2];
  const float* emb   = (const float*)d_in[3];
  const float* W1    = (const float*)d_in[4];
  const float* root1 = (const float*)d_in[5];
  const float* bias1 = (const float*)d_in[6];
  const float* W2    = (const float*)d_in[7];
  const float* root2 = (const float*)d_in[8];
  const float* bias2 = (const float*)d_in[9];
  float* out = (float*)d_out;

  const int N = in_sizes[0];
  const int E = in_sizes[2];
  const int* esrc = eidx;
  const int* edst = eidx + E;

  // workspace: h0 | h1 | counts
  char* ws = (char*)d_ws;
  float*    h0     = (float*)ws;
  float*    h1     = (float*)(ws + (size_t)N * EMB_DIM * sizeof(float));
  unsigned* counts = (unsigned*)(ws + 2 * (size_t)N * EMB_DIM * sizeof(float));

  const int TPB = 256;
  const int nh      = N * EMB_DIM;                 // 3.2M
  const int ncounts = N * NUM_RELS;                // 8M
  const int gemm_blocks = (((N + 15) / 16) + 7) / 8;   // 8 waves/block
  const int edge_blocks = 4096;                    // persistent, amortize LDS fill

  // h0 = emb[x]
  gather_rows_kernel<<<(nh + TPB - 1) / TPB, TPB, 0, stream>>>(emb, x, h0, N);

  // counts (shared across both layers: same edges + types)
  zero_u32_kernel<<<(ncounts + TPB - 1) / TPB, TPB, 0, stream>>>(counts, ncounts);
  count_edges_kernel<<<(E + TPB - 1) / TPB, TPB, 0, stream>>>(edst, etype, counts, E);

  // ---- layer 1: h1 = relu( h0@root1 + bias1 + agg1 ) ----
  root_gemm_wmma_kernel<<<gemm_blocks, TPB, 0, stream>>>(h0, root1, bias1, h1, N);
  edge_msg_kernel<<<edge_blocks, TPB, 0, stream>>>(h0, W1, counts, esrc, edst, etype, h1, E);
  relu_kernel<<<(nh + TPB - 1) / TPB, TPB, 0, stream>>>(h1, nh);

  // ---- layer 2: out = h1@root2 + bias2 + agg2 ----
  root_gemm_wmma_kernel<<<gemm_blocks, TPB, 0, stream>>>(h1, root2, bias2, out, N);
  edge_msg_kernel<<<edge_blocks, TPB, 0, stream>>>(h1, W2, counts, esrc, edst, etype, out, E);
}